// hierarchical_sae_criterion_28372553957682
// MI455X (gfx1250) — compile-verified
//
#include <hip/hip_runtime.h>
#include <math.h>

typedef float v2f __attribute__((ext_vector_type(2)));
typedef float v4f __attribute__((ext_vector_type(4)));
typedef float v8f __attribute__((ext_vector_type(8)));

#define NROW 4096   // n samples
#define DC   4096   // common feature dim
#define NB   4      // blocks
#define DB   2048   // block feature dim
#define NP   5      // cox problems: pred + 4 block hazards
#define SSE_BLOCKS 512
#define COX_WGS 32  // 32 workgroups * 8 waves * 16 rows = 4096 rows

// ---------------------------------------------------------------------------
// Kernel 1: streaming weighted-SSE partial sums (bandwidth bound, ~384MB).
// grid = (SSE_BLOCKS, 5): y==0 -> common, y==1..4 -> block b = y-1.
// Deterministic: each block writes its own partial (no atomics).
// ---------------------------------------------------------------------------
__global__ void sse_kernel(const float* __restrict__ ctar,
                           const float* __restrict__ crec,
                           const float* __restrict__ btar,
                           const float* __restrict__ brec,
                           const float* __restrict__ sw,
                           float* __restrict__ partials) {
  __shared__ float red[256];
  const int sec = blockIdx.y;
  const v4f* rp;
  const v4f* tp;
  int nq, rsh;
  if (sec == 0) {
    rp = (const v4f*)crec;
    tp = (const v4f*)ctar;
    nq = (NROW * DC) / 4;   // 4,194,304 float4
    rsh = 10;               // 1024 float4 per row
  } else {
    const int b = sec - 1;
    rp = (const v4f*)(brec + (size_t)b * NROW * DB);
    tp = (const v4f*)(btar + (size_t)b * NROW * DB);
    nq = (NROW * DB) / 4;   // 2,097,152 float4
    rsh = 9;                // 512 float4 per row
  }
  float acc = 0.0f;
  const int stride = gridDim.x * blockDim.x;
  for (int q = blockIdx.x * blockDim.x + threadIdx.x; q < nq; q += stride) {
    v4f r = __builtin_nontemporal_load(&rp[q]);   // read-once stream: NT hint
    v4f t = __builtin_nontemporal_load(&tp[q]);
    float w = sw[q >> rsh];
    v4f d = r - t;
    acc = fmaf(w, d.x * d.x + d.y * d.y + d.z * d.z + d.w * d.w, acc);
  }
  red[threadIdx.x] = acc;
  __syncthreads();
  for (int s = 128; s > 0; s >>= 1) {
    if (threadIdx.x < s) red[threadIdx.x] += red[threadIdx.x + s];
    __syncthreads();
  }
  if (threadIdx.x == 0) partials[sec * gridDim.x + blockIdx.x] = red[0];
}

// ---------------------------------------------------------------------------
// Kernel 2: batched Cox risk-set mat-vec via V_WMMA_F32_16X16X4_F32.
//   risk[i][n] = sum_j (t[j] >= t[i]) * exp(P_n[j]),  n = 0..4 (pred, hz0..3)
// A tile (16x4 f32): indicator built on the fly.
//   lanes 0-15 : row m=lane,    VGPR0=K0, VGPR1=K1
//   lanes 16-31: row m=lane-16, VGPR0=K2, VGPR1=K3
// B tile (4x16 f32): column n = problem index (cols 5..15 never read from C).
// C/D (16x16 f32, v8f): VGPR r -> (M=r, N=lane) / (M=r+8, N=lane-16).
// 8 waves per WG, one 16-row i-tile per wave; exp matrix staged in LDS.
// ---------------------------------------------------------------------------
__global__ void cox_wmma_kernel(const float* __restrict__ pred,
                                const float* __restrict__ hz,
                                const float* __restrict__ sw,
                                const int* __restrict__ st,
                                const int* __restrict__ ev,
                                float* __restrict__ spart) {
  extern __shared__ char smem[];
  float* E = (float*)smem;                                    // NP * NROW f32 (80 KB)
  int*   T = (int*)(smem + NP * NROW * sizeof(float));        // NROW i32 (16 KB)
  float* S = (float*)(smem + NP * NROW * sizeof(float) + NROW * sizeof(int)); // 256 f32

  const int tid = threadIdx.x;
  for (int idx = tid; idx < NP * NROW; idx += blockDim.x) {
    const int n = idx >> 12;
    const int j = idx & (NROW - 1);
    const float p = (n == 0) ? pred[j] : hz[(n - 1) * NROW + j];
    E[idx] = expf(p);
  }
  for (int j = tid; j < NROW; j += blockDim.x) T[j] = st[j];
  __syncthreads();

  const int lane = tid & 31;
  const int wave = tid >> 5;
  const int itile = blockIdx.x * 8 + wave;
  const int m = lane & 15;                 // row within tile (A), column n (B/C)
  const int ti = T[itile * 16 + m];
  const int n2 = (m < NP) ? m : 0;         // clamp: cols >= NP compute unused garbage
  const int kb = (lane < 16) ? 0 : 2;
  const float* Erow = E + n2 * NROW;

  v8f c = {};
  for (int jb = 0; jb < NROW; jb += 4) {
    const int jk = jb + kb;
    v2f a, b;
    a.x = (T[jk]     >= ti) ? 1.0f : 0.0f;
    a.y = (T[jk + 1] >= ti) ? 1.0f : 0.0f;
    b.x = Erow[jk];
    b.y = Erow[jk + 1];
    c = __builtin_amdgcn_wmma_f32_16x16x4_f32(false, a, false, b, (short)0, c,
                                              false, false);
  }

  // s_n = sum_i w[i] * e[i] * (P_n[i] - log(risk[i][n]))  (lane handles col n=m)
  float s_lane = 0.0f;
  if (m < NP) {
    const float* P = (m == 0) ? pred : hz + (m - 1) * NROW;
    const int rowbase = itile * 16 + ((lane < 16) ? 0 : 8);
#pragma unroll
    for (int r = 0; r < 8; ++r) {
      const int i = rowbase + r;
      const float diff = P[i] - logf(c[r]);
      s_lane = fmaf(sw[i] * (float)ev[i], diff, s_lane);
    }
  }
  S[tid] = s_lane;
  __syncthreads();
  if (tid < NP) {
    float acc = 0.0f;
    for (int w = 0; w < 8; ++w) acc += S[w * 32 + tid] + S[w * 32 + 16 + tid];
    spart[blockIdx.x * NP + tid] = acc;   // deterministic per-WG partial
  }
}

// ---------------------------------------------------------------------------
// Kernel 3: finalize — fixed-order sums of partials, n_obs, write 10 outputs.
// out = [cox(pred), common, block_rec[0..3], block_haz[0..3]]
// ---------------------------------------------------------------------------
__device__ __forceinline__ float block_reduce(float v, float* red) {
  const int tid = threadIdx.x;
  red[tid] = v;
  __syncthreads();
  for (int s = 128; s > 0; s >>= 1) {
    if (tid < s) red[tid] += red[tid + s];
    __syncthreads();
  }
  const float r = red[0];
  __syncthreads();
  return r;
}

__global__ void finalize_kernel(const float* __restrict__ ssep,
                                const float* __restrict__ coxp,
                                const int* __restrict__ ev,
                                float* __restrict__ out) {
  __shared__ float red[256];
  const int tid = threadIdx.x;

  float nacc = 0.0f;
  for (int i = tid; i < NROW; i += 256) nacc += (float)ev[i];
  const float nobs = block_reduce(nacc, red);

  for (int sec = 0; sec < 5; ++sec) {
    float a = 0.0f;
    for (int i = tid; i < SSE_BLOCKS; i += 256) a += ssep[sec * SSE_BLOCKS + i];
    const float tot = block_reduce(a, red);
    if (tid == 0) {
      if (sec == 0) out[1] = tot / (float)DC;
      else          out[2 + (sec - 1)] = tot / (float)DB;
    }
  }

  if (tid < NP) {
    float s = 0.0f;
    for (int wg = 0; wg < COX_WGS; ++wg) s += coxp[wg * NP + tid];
    const float loss = (nobs > 0.0f) ? (-s / fmaxf(nobs, 1.0f)) : 0.0f;
    if (tid == 0) out[0] = loss;
    else          out[6 + (tid - 1)] = loss;
  }
}

// ---------------------------------------------------------------------------
extern "C" void kernel_launch(void* const* d_in, const int* in_sizes, int n_in,
                              void* d_out, int out_size, void* d_ws, size_t ws_size,
                              hipStream_t stream) {
  (void)in_sizes; (void)n_in; (void)out_size; (void)ws_size;
  const float* pred = (const float*)d_in[0];
  const float* ctar = (const float*)d_in[1];
  const float* crec = (const float*)d_in[2];
  const float* btar = (const float*)d_in[3];
  const float* brec = (const float*)d_in[4];
  const float* bhz  = (const float*)d_in[5];
  const float* sw   = (const float*)d_in[6];
  const int*   st   = (const int*)d_in[7];
  const int*   ev   = (const int*)d_in[8];
  float* out = (float*)d_out;
  float* ws  = (float*)d_ws;

  float* ssep = ws;                       // [5][SSE_BLOCKS]
  float* coxp = ws + 5 * SSE_BLOCKS;      // [COX_WGS][NP]

  sse_kernel<<<dim3(SSE_BLOCKS, 5), 256, 0, stream>>>(ctar, crec, btar, brec, sw, ssep);

  const size_t lds = (size_t)NP * NROW * sizeof(float) + NROW * sizeof(int) +
                     256 * sizeof(float);  // 99,328 B (< 320 KB WGP LDS)
  cox_wmma_kernel<<<COX_WGS, 256, lds, stream>>>(pred, bhz, sw, st, ev, coxp);

  finalize_kernel<<<1, 256, 0, stream>>>(ssep, coxp, ev, out);
}